// EvolveGCNO_RecurrentGCN_16192026706534
// MI455X (gfx1250) — compile-verified
//
#include <hip/hip_runtime.h>
#include <hip/hip_bf16.h>
#include <math.h>

#define C_DIM   128
#define NNODES  50000
#define NEDGES  600000

typedef float v2f __attribute__((ext_vector_type(2)));
typedef float v8f __attribute__((ext_vector_type(8)));

__device__ __forceinline__ float sigmoidf_(float x) { return 1.0f / (1.0f + __expf(-x)); }

// ---------------------------------------------------------------------------
// 1) EvolveGCNO GRU step on the weight matrix: W = GRU(W0, W0).
//    Writes the evolved weight TRANSPOSED (Wt[n][k] = W[k][n]) so the GEMM's
//    B operand pairs (k, k+1) are contiguous -> single global_load_b64.
//    grid = 128 (row r), block = 128 (col c). W0 row staged in LDS.
// ---------------------------------------------------------------------------
__global__ void __launch_bounds__(C_DIM) gru_evolve(
    const float* __restrict__ W0, const float* __restrict__ wih,
    const float* __restrict__ whh, const float* __restrict__ bih,
    const float* __restrict__ bhh, float* __restrict__ Wt)
{
    __shared__ float row[C_DIM];
    const int r = blockIdx.x;
    const int c = threadIdx.x;
    row[c] = W0[r * C_DIM + c];
    __syncthreads();

    const float* wr = wih + (size_t)c * C_DIM;
    const float* wz = wih + (size_t)(C_DIM + c) * C_DIM;
    const float* wn = wih + (size_t)(2 * C_DIM + c) * C_DIM;
    const float* hr = whh + (size_t)c * C_DIM;
    const float* hz = whh + (size_t)(C_DIM + c) * C_DIM;
    const float* hn = whh + (size_t)(2 * C_DIM + c) * C_DIM;

    float axr = 0.f, axz = 0.f, axn = 0.f, ahr = 0.f, ahz = 0.f, ahn = 0.f;
#pragma unroll 4
    for (int k = 0; k < C_DIM; ++k) {
        const float w0 = row[k];
        axr = fmaf(w0, wr[k], axr);
        axz = fmaf(w0, wz[k], axz);
        axn = fmaf(w0, wn[k], axn);
        ahr = fmaf(w0, hr[k], ahr);
        ahz = fmaf(w0, hz[k], ahz);
        ahn = fmaf(w0, hn[k], ahn);
    }
    const float rg = sigmoidf_(axr + bih[c] + ahr + bhh[c]);
    const float zg = sigmoidf_(axz + bih[C_DIM + c] + ahz + bhh[C_DIM + c]);
    const float ng = tanhf(axn + bih[2 * C_DIM + c] + rg * (ahn + bhh[2 * C_DIM + c]));
    // transposed store: Wt[c][r] = W[r][c]
    Wt[(size_t)c * C_DIM + r] = (1.0f - zg) * ng + zg * row[c];
}

// ---------------------------------------------------------------------------
// 2) Degree (with self-loop weight 1.0) and symmetric norm factor
// ---------------------------------------------------------------------------
__global__ void deg_init(float* __restrict__ deg)
{
    const int i = blockIdx.x * blockDim.x + threadIdx.x;
    if (i < NNODES) deg[i] = 1.0f;  // self-loop contribution
}

__global__ void deg_accum(const int* __restrict__ dst, const float* __restrict__ ew,
                          float* __restrict__ deg)
{
    const int e = blockIdx.x * blockDim.x + threadIdx.x;
    if (e < NEDGES)
        __hip_atomic_fetch_add(&deg[dst[e]], ew[e],
                               __ATOMIC_RELAXED, __HIP_MEMORY_SCOPE_AGENT);
}

__global__ void dinv_kernel(const float* __restrict__ deg, float* __restrict__ dinv)
{
    const int i = blockIdx.x * blockDim.x + threadIdx.x;
    if (i < NNODES) {
        const float d = deg[i];
        dinv[i] = (d > 0.0f) ? rsqrtf(d) : 0.0f;
    }
}

// ---------------------------------------------------------------------------
// 3) xw = x @ W using V_WMMA_F32_16X16X4_F32 (full fp32 precision).
//    One block = 8 waves; block handles a 16-row x 128-col output stripe.
//    Wave w computes the 16x16 tile at columns [16w, 16w+16).
//    A comes from the LDS-staged x stripe (ds_load_2addr_b64);
//    B comes from transposed W (one global_load_b64 per k-step).
// ---------------------------------------------------------------------------
__global__ void __launch_bounds__(256) gemm_xw(
    const float* __restrict__ X, const float* __restrict__ Wt,
    float* __restrict__ XW)
{
    __shared__ float Xs[16][C_DIM];   // 8 KB stripe of x, shared by all 8 waves
    const int m0 = blockIdx.x * 16;
    const int t  = threadIdx.x;

    // cooperative load: 256 threads x 8 floats = 16*128
#pragma unroll
    for (int j = 0; j < 8; ++j) {
        const int i = t * 8 + j;
        Xs[i >> 7][i & 127] = X[(size_t)(m0 + (i >> 7)) * C_DIM + (i & 127)];
    }
    __syncthreads();

    const int lane = t & 31;
    const int li   = lane & 15;
    const int half = lane >> 4;
    const int n0   = (t >> 5) * 16;

    const float* wcol = Wt + (size_t)(n0 + li) * C_DIM;  // column (n0+li) of W

    v8f acc = {};
#pragma unroll
    for (int k0 = 0; k0 < C_DIM; k0 += 4) {
        const int ka = k0 + 2 * half;
        v2f a, b;
        a.x = Xs[li][ka];
        a.y = Xs[li][ka + 1];
        b   = *(const v2f*)(wcol + ka);   // {W[ka][n], W[ka+1][n]} contiguous
        acc = __builtin_amdgcn_wmma_f32_16x16x4_f32(
            /*neg_a=*/false, a, /*neg_b=*/false, b,
            /*c_mod=*/(short)0, acc, /*reuse_a=*/false, /*reuse_b=*/false);
    }
#pragma unroll
    for (int v = 0; v < 8; ++v)
        XW[(size_t)(m0 + v + 8 * half) * C_DIM + n0 + li] = acc[v];
}

// ---------------------------------------------------------------------------
// 4) h init with self-loop term: h[i,:] = dinv[i]^2 * xw[i,:]
// ---------------------------------------------------------------------------
__global__ void h_init(const float* __restrict__ dinv, const float* __restrict__ xw,
                       float* __restrict__ h)
{
    const int i = blockIdx.x * blockDim.x + threadIdx.x;   // over N*C
    if (i < NNODES * C_DIM) {
        const float di = dinv[i >> 7];
        h[i] = di * di * xw[i];
    }
}

// ---------------------------------------------------------------------------
// 5) Edge gather-scatter: h[dst] += norm * xw[src].  One wave32 per edge,
//    each lane owns a float4 (32*4 = 128 channels). xw/h fit in the 192MB L2,
//    so the random traffic stays on-chip.
// ---------------------------------------------------------------------------
__global__ void __launch_bounds__(256) edge_scatter(
    const int* __restrict__ src, const int* __restrict__ dst,
    const float* __restrict__ ew, const float* __restrict__ dinv,
    const float* __restrict__ xw, float* __restrict__ h)
{
    const int gid  = blockIdx.x * blockDim.x + threadIdx.x;
    const int e    = gid >> 5;
    const int lane = gid & 31;
    if (e >= NEDGES) return;

    const int s = src[e];
    const int d = dst[e];
    const float norm = dinv[s] * ew[e] * dinv[d];

    const int c = lane * 4;
    const float4 v = *(const float4*)(xw + (size_t)s * C_DIM + c);
    float* hp = h + (size_t)d * C_DIM + c;
    __hip_atomic_fetch_add(hp + 0, norm * v.x, __ATOMIC_RELAXED, __HIP_MEMORY_SCOPE_AGENT);
    __hip_atomic_fetch_add(hp + 1, norm * v.y, __ATOMIC_RELAXED, __HIP_MEMORY_SCOPE_AGENT);
    __hip_atomic_fetch_add(hp + 2, norm * v.z, __ATOMIC_RELAXED, __HIP_MEMORY_SCOPE_AGENT);
    __hip_atomic_fetch_add(hp + 3, norm * v.w, __ATOMIC_RELAXED, __HIP_MEMORY_SCOPE_AGENT);
}

// ---------------------------------------------------------------------------
// 6) out[i] = relu(h[i,:]) . lin_w + lin_b   (one wave32 per node)
// ---------------------------------------------------------------------------
__global__ void __launch_bounds__(256) final_linear(
    const float* __restrict__ h, const float* __restrict__ lw,
    const float* __restrict__ lb, float* __restrict__ out)
{
    const int gid  = blockIdx.x * blockDim.x + threadIdx.x;
    const int node = gid >> 5;
    const int lane = gid & 31;
    if (node >= NNODES) return;

    const int c = lane * 4;
    const float4 hv = *(const float4*)(h + (size_t)node * C_DIM + c);
    const float4 wv = *(const float4*)(lw + c);
    float acc = fmaxf(hv.x, 0.f) * wv.x + fmaxf(hv.y, 0.f) * wv.y +
                fmaxf(hv.z, 0.f) * wv.z + fmaxf(hv.w, 0.f) * wv.w;
#pragma unroll
    for (int off = 16; off > 0; off >>= 1)
        acc += __shfl_down(acc, off, 32);
    if (lane == 0) out[node] = acc + lb[0];
}

// ---------------------------------------------------------------------------
extern "C" void kernel_launch(void* const* d_in, const int* in_sizes, int n_in,
                              void* d_out, int out_size, void* d_ws, size_t ws_size,
                              hipStream_t stream)
{
    const float* x   = (const float*)d_in[0];
    const int*   ei  = (const int*)d_in[1];     // [2, E] flat: src then dst
    const float* ew  = (const float*)d_in[2];
    const float* W0  = (const float*)d_in[3];
    const float* wih = (const float*)d_in[4];
    const float* whh = (const float*)d_in[5];
    const float* bih = (const float*)d_in[6];
    const float* bhh = (const float*)d_in[7];
    const float* lw  = (const float*)d_in[8];
    const float* lb  = (const float*)d_in[9];
    float* out = (float*)d_out;

    char* ws = (char*)d_ws;
    float* Wt   = (float*)(ws);                                    //   64 KB (transposed W)
    float* deg  = (float*)(ws + (64 << 10));                       //  256 KB slot
    float* dinv = (float*)(ws + (64 << 10) + (256 << 10));         //  256 KB slot
    float* xw   = (float*)(ws + (64 << 10) + (512 << 10));         // 25.6 MB
    float* h    = (float*)(ws + (64 << 10) + (512 << 10) + (size_t)NNODES * C_DIM * 4);

    const int* src = ei;
    const int* dst = ei + NEDGES;

    gru_evolve<<<C_DIM, C_DIM, 0, stream>>>(W0, wih, whh, bih, bhh, Wt);
    deg_init<<<(NNODES + 255) / 256, 256, 0, stream>>>(deg);
    deg_accum<<<(NEDGES + 255) / 256, 256, 0, stream>>>(dst, ew, deg);
    dinv_kernel<<<(NNODES + 255) / 256, 256, 0, stream>>>(deg, dinv);
    gemm_xw<<<NNODES / 16, 256, 0, stream>>>(x, Wt, xw);
    h_init<<<(NNODES * C_DIM + 255) / 256, 256, 0, stream>>>(dinv, xw, h);
    edge_scatter<<<((size_t)NEDGES * 32 + 255) / 256, 256, 0, stream>>>(src, dst, ew, dinv, xw, h);
    final_linear<<<((size_t)NNODES * 32 + 255) / 256, 256, 0, stream>>>(h, lw, lb, out);
}